// JacobiConv_1589137899752
// MI455X (gfx1250) — compile-verified
//
#include <hip/hip_runtime.h>

typedef __attribute__((ext_vector_type(2))) float v2f;
typedef __attribute__((ext_vector_type(8))) float v8f;
typedef __attribute__((ext_vector_type(4))) int   v4i;

#define NN   16384      // nodes
#define CC   32         // channels
#define KORD 10         // polynomial order
#define KC   64         // K-chunk per LDS stage
#define APAD 68         // padded LDS row stride (floats) -> conflict-free b64 reads
#define AJ   1.0f       // jacobi a
#define BJ   1.0f       // jacobi b

#define ATILE (8 * 16 * APAD)   // floats per A buffer (8 waves x 16 rows x APAD)
#define BTILE (CC * APAD)       // floats per B buffer (transposed, padded)

#if __has_builtin(__builtin_amdgcn_global_load_async_to_lds_b128) && \
    __has_builtin(__builtin_amdgcn_global_load_async_to_lds_b32)
#define USE_ASYNC 1
#else
#define USE_ASYNC 0
#endif

__device__ __forceinline__ void wait_async0() {
#if USE_ASYNC
#if __has_builtin(__builtin_amdgcn_s_wait_asynccnt)
    __builtin_amdgcn_s_wait_asynccnt(0);
#else
    asm volatile("s_wait_asynccnt 0" ::: "memory");
#endif
#endif
}

// ---------------------------------------------------------------- init: y = comb[0] * x
__global__ __launch_bounds__(256) void jacobi_init(const float* __restrict__ x,
                                                   const float* __restrict__ comb,
                                                   float* __restrict__ y) {
    int idx = blockIdx.x * 256 + threadIdx.x;
    y[idx] = comb[idx & (CC - 1)] * x[idx];
}

// ---------------------------------------------------------------- one recurrence step:
//   T = L @ Pprev        (WMMA f32 16x16x4 GEMM, async double-buffered staging)
//   Pout = cL*T + cP*Pprev - cPP*Ppp
//   y   += comb[i] * Pout
__global__ __launch_bounds__(256) void jacobi_step(
    const float* __restrict__ L,
    const float* __restrict__ Pprev,
    const float* __restrict__ Ppp,      // null for i==1
    float*       __restrict__ Pout,
    float*       __restrict__ y,
    const float* __restrict__ conv,
    const float* __restrict__ comb,
    int i)
{
    __shared__ float sA[2][ATILE];      // per-wave 16 x KC tiles, padded rows
    __shared__ float sBt[2][BTILE];     // B chunk transposed: [n][k], padded rows

    const int tid   = threadIdx.x;
    const int wave  = tid >> 5;
    const int lane  = tid & 31;
    const int m0    = blockIdx.x * 128 + wave * 16;   // this wave's first output row
    const int mlane = lane & 15;
    const int khalf = (lane >> 4) << 1;               // 0 or 2 (A/B fragment K pair)

    // ---- staging helpers -------------------------------------------------
    auto stageA = [&](int kk, int buf) {
        const int c4 = lane & 15;                     // float4 column index 0..15
#pragma unroll
        for (int s = 0; s < 8; ++s) {
            const int row = s * 2 + (lane >> 4);      // 0..15
            const float* gp = L + (size_t)(m0 + row) * NN + kk + c4 * 4;
            float* lp = &sA[buf][(wave * 16 + row) * APAD + c4 * 4];
#if USE_ASYNC
            __builtin_amdgcn_global_load_async_to_lds_b128((v4i*)gp, (v4i*)lp, 0, 0);
#else
            *(float4*)lp = *(const float4*)gp;
#endif
        }
    };
    auto stageB = [&](int kk, int buf) {
#pragma unroll
        for (int s = 0; s < 8; ++s) {
            const int idx = s * 256 + tid;            // 0..2047
            const int bk  = idx >> 5;                 // 0..63
            const int bn  = idx & 31;                 // 0..31
            const float* gp = Pprev + (size_t)(kk + bk) * CC + bn;
            float* lp = &sBt[buf][bn * APAD + bk];
#if USE_ASYNC
            __builtin_amdgcn_global_load_async_to_lds_b32((int*)gp, (int*)lp, 0, 0);
#else
            *lp = *gp;
#endif
        }
    };

    v8f acc0 = {};   // cols 0..15
    v8f acc1 = {};   // cols 16..31

    // ---- prologue: stage first chunk ------------------------------------
    stageA(0, 0);
    stageB(0, 0);
    wait_async0();
    __syncthreads();

    int cur = 0;
    for (int kk = 0; kk < NN; kk += KC) {
        const int nxt = cur ^ 1;
        // ---- kick off async staging of the next chunk while computing ---
        if (kk + KC < NN) {
            stageA(kk + KC, nxt);
            stageB(kk + KC, nxt);
        }

        // ---- 16 WMMA steps of K=4 each on current buffer ----------------
        const float* aRow  = &sA[cur][(wave * 16 + mlane) * APAD];
        const float* b0Row = &sBt[cur][mlane * APAD];
        const float* b1Row = &sBt[cur][(mlane + 16) * APAD];
#pragma unroll
        for (int k4 = 0; k4 < KC / 4; ++k4) {
            const int k0 = k4 * 4 + khalf;
            v2f a  = *(const v2f*)(aRow  + k0);
            v2f b0 = *(const v2f*)(b0Row + k0);
            v2f b1 = *(const v2f*)(b1Row + k0);
            acc0 = __builtin_amdgcn_wmma_f32_16x16x4_f32(
                false, a, false, b0, (short)0, acc0, false, false);
            acc1 = __builtin_amdgcn_wmma_f32_16x16x4_f32(
                false, a, false, b1, (short)0, acc1, false, false);
        }

        // ---- next buffer must be fully landed before anyone reads it ----
        wait_async0();
        __syncthreads();
        cur = nxt;
    }

    // ---- Jacobi recurrence coefficients (scalar, uniform) ---------------
    float cL, cP, cPP;
    if (i == 1) {
        const float al0 = 2.0f * tanhf(conv[0]);
        cL  = al0 * (AJ + BJ + 2.0f) * 0.5f;
        cP  = al0 * (AJ - BJ) * 0.5f;
        cPP = 0.0f;
    } else {
        const float fi  = (float)i;
        const float ab  = AJ + BJ;
        const float div = 2.0f * fi * (fi + ab) * (2.0f * fi + ab - 2.0f);
        const float r1  = (2.0f * fi + ab) * (2.0f * fi + ab - 1.0f) * (2.0f * fi + ab - 2.0f) / div;
        const float r2  = (2.0f * fi + ab - 1.0f) * (AJ * AJ - BJ * BJ) / div;
        const float r3  = 2.0f * (fi + AJ - 1.0f) * (fi + BJ - 1.0f) * (2.0f * fi + ab) / div;
        const float al1 = 2.0f * tanhf(conv[i - 1]);
        const float al2 = 2.0f * tanhf(conv[i - 2]);
        cL  = al1 * r1;
        cP  = al1 * r2;
        cPP = al1 * al2 * r3;
    }

    // ---- epilogue: recurrence + output accumulation ---------------------
    // C/D layout: VGPR j -> rows j (lanes 0-15) / j+8 (lanes 16-31)
    const int n0 = mlane;
    const float cb0 = comb[i * CC + n0];
    const float cb1 = comb[i * CC + n0 + 16];
#pragma unroll
    for (int j = 0; j < 8; ++j) {
        const int row = m0 + j + ((lane >> 4) << 3);
        const size_t off0 = (size_t)row * CC + n0;
        const size_t off1 = off0 + 16;
        const float p0 = Pprev[off0];
        const float p1 = Pprev[off1];
        float q0 = 0.0f, q1 = 0.0f;
        if (Ppp) { q0 = Ppp[off0]; q1 = Ppp[off1]; }
        const float v0 = cL * acc0[j] + cP * p0 - cPP * q0;
        const float v1 = cL * acc1[j] + cP * p1 - cPP * q1;
        Pout[off0] = v0;
        Pout[off1] = v1;
        y[off0] += cb0 * v0;
        y[off1] += cb1 * v1;
    }
}

// ----------------------------------------------------------------
extern "C" void kernel_launch(void* const* d_in, const int* in_sizes, int n_in,
                              void* d_out, int out_size, void* d_ws, size_t ws_size,
                              hipStream_t stream) {
    (void)in_sizes; (void)n_in; (void)out_size; (void)ws_size;
    const float* x    = (const float*)d_in[0];
    const float* L    = (const float*)d_in[1];
    const float* conv = (const float*)d_in[2];
    const float* comb = (const float*)d_in[3];
    float* y = (float*)d_out;

    float* P[3];
    P[0] = (float*)d_ws;
    P[1] = P[0] + (size_t)NN * CC;
    P[2] = P[1] + (size_t)NN * CC;

    jacobi_init<<<(NN * CC) / 256, 256, 0, stream>>>(x, comb, y);

    const float* prev = x;
    const float* pp   = nullptr;
    for (int i = 1; i <= KORD; ++i) {
        float* out = P[(i - 1) % 3];
        jacobi_step<<<NN / 128, 256, 0, stream>>>(L, prev, pp, out, y, conv, comb, i);
        pp = prev;
        prev = out;
    }
}